// MAGNNIntraMetapathAggregator_62414464745715
// MI455X (gfx1250) — compile-verified
//
#include <hip/hip_runtime.h>
#include <math.h>

// ---------------------------------------------------------------------------
// MAGNN intra-metapath aggregator for MI455X (gfx1250, wave32).
//   reps = mean_L(paths)                      (streams 819 MB once -> ~35us floor)
//   e    = leakyrelu(a_t.target + a_r @ reps.T)
//   out  = softmax_n(e) @ reps                (online softmax, deterministic merge)
// Logit tiles via V_WMMA_F32_16X16X4_F32, A staged zero-padded in LDS,
// dual accumulator chains, non-temporal streaming loads for paths.
// ---------------------------------------------------------------------------

typedef __attribute__((ext_vector_type(2))) float v2f;
typedef __attribute__((ext_vector_type(4))) float v4f;
typedef __attribute__((ext_vector_type(8))) float v8f;

#define DIMD   128
#define HEADS  8
#define NPATH  100000
#define LROWS  16
#define TILE   16
#define NTILES (NPATH / TILE)   // 6250 exactly
#define NBLK   640
#define NEG_BIG (-3.0e38f)

__global__ __launch_bounds__(256)
void magnn_pass1(const float* __restrict__ tgt,      // (128)
                 const float* __restrict__ paths,    // (N,16,128)
                 const float* __restrict__ afc,      // (8,256)
                 float* __restrict__ pm,             // (NBLK,8)  running max
                 float* __restrict__ ps,             // (NBLK,8)  running sumexp
                 float* __restrict__ pacc)           // (NBLK,8,128) weighted sums
{
    __shared__ float lds_reps[TILE * DIMD];   // 8 KB : 16 path means
    __shared__ float lds_a[TILE * DIMD];      // 8 KB : a_r zero-padded to 16 rows
    __shared__ float lds_e[HEADS * TILE];     // logits tile
    __shared__ float lds_at[HEADS];           // a_t . target

    const int t    = threadIdx.x;
    const int wave = t >> 5;          // 0..7  -> owns head `wave`
    const int lane = t & 31;          // 0..31 -> owns columns lane*4 .. +3

    // ---- one-time: a_t.target and zero-padded A = a_r (heads 8..15 = 0) ----
    if (t < HEADS) {
        float s = 0.f;
        for (int d = 0; d < DIMD; ++d) s += afc[t * (2 * DIMD) + d] * tgt[d];
        lds_at[t] = s;
    }
    #pragma unroll
    for (int i = 0; i < (TILE * DIMD) / 256; ++i) {
        const int idx = i * 256 + t;
        const int r   = idx >> 7;            // 0..15
        const int d   = idx & (DIMD - 1);
        lds_a[idx] = (r < HEADS) ? afc[r * (2 * DIMD) + DIMD + d] : 0.f;
    }

    // per-thread online-softmax state (head = wave, 4 columns per lane)
    float m    = NEG_BIG;
    float ssum = 0.f;
    v4f   acc  = (v4f)(0.f);

    for (int tile = blockIdx.x; tile < NTILES; tile += gridDim.x) {
        __syncthreads();   // protect LDS reuse across iterations (and init fills)

        // ---- stage 1: stream 16 paths (NT loads), reduce over L -----------
        const int n0 = tile * TILE;
        #pragma unroll
        for (int j = 0; j < 2; ++j) {
            const int r = wave * 2 + j;                       // path row 0..15
            const float* p = paths + (size_t)(n0 + r) * (LROWS * DIMD) + lane * 4;
            v4f s4 = (v4f)(0.f);
            #pragma unroll
            for (int l = 0; l < LROWS; ++l) {
                const v4f v = __builtin_nontemporal_load((const v4f*)(p + l * DIMD));
                s4 += v;
            }
            s4 *= (1.0f / (float)LROWS);
            *(v4f*)&lds_reps[r * DIMD + lane * 4] = s4;
        }

        // prefetch next tile's data (emits global_prefetch_b8)
        const int ntile = tile + gridDim.x;
        if (ntile < NTILES) {
            const char* np = (const char*)(paths + (size_t)ntile * TILE * LROWS * DIMD);
            __builtin_prefetch(np + t * 512, 0, 0);
        }
        __syncthreads();

        // ---- stage 2: wave 0 computes e[h][n] with WMMA f32 16x16x4 -------
        if (wave == 0) {
            v8f c0 = {}, c1 = {};
            const int r16   = lane & 15;     // row index (A: head, B: path)
            const int koff  = (lane >> 4) * 2;  // low lanes K{0,1}, high K{2,3}
            const float* As = &lds_a[r16 * DIMD + koff];
            const float* Bs = &lds_reps[r16 * DIMD + koff];
            #pragma unroll 4
            for (int k0 = 0; k0 < DIMD; k0 += 8) {
                const v2f a0 = *(const v2f*)(As + k0);
                const v2f b0 = *(const v2f*)(Bs + k0);
                c0 = __builtin_amdgcn_wmma_f32_16x16x4_f32(
                        false, a0, false, b0, (short)0, c0, false, false);
                const v2f a1 = *(const v2f*)(As + k0 + 4);
                const v2f b1 = *(const v2f*)(Bs + k0 + 4);
                c1 = __builtin_amdgcn_wmma_f32_16x16x4_f32(
                        false, a1, false, b1, (short)0, c1, false, false);
            }
            // C/D layout: VGPR v, lanes 0..15 -> row v (heads 0..7), N = lane
            if (lane < 16) {
                #pragma unroll
                for (int v = 0; v < 8; ++v) {
                    float e = lds_at[v] + c0[v] + c1[v];
                    e = (e > 0.f) ? e : 0.2f * e;        // LeakyReLU(0.2)
                    lds_e[v * TILE + lane] = e;
                }
            }
        }
        __syncthreads();

        // ---- stage 3: online softmax accumulate (head = wave) -------------
        float tmax = lds_e[wave * TILE];
        #pragma unroll
        for (int n = 1; n < TILE; ++n) tmax = fmaxf(tmax, lds_e[wave * TILE + n]);
        const float mnew  = fmaxf(m, tmax);
        const float scale = __expf(m - mnew);            // exp(-inf)=0 first time
        ssum *= scale;
        acc  *= scale;
        #pragma unroll
        for (int n = 0; n < TILE; ++n) {
            const float w  = __expf(lds_e[wave * TILE + n] - mnew);
            const v4f   r4 = *(const v4f*)&lds_reps[n * DIMD + lane * 4];
            ssum += w;
            acc  += w * r4;
        }
        m = mnew;
    }

    // ---- write per-block partials -----------------------------------------
    if (lane == 0) {
        pm[blockIdx.x * HEADS + wave] = m;
        ps[blockIdx.x * HEADS + wave] = ssum;
    }
    *(v4f*)&pacc[(size_t)blockIdx.x * (HEADS * DIMD) + wave * DIMD + lane * 4] = acc;
}

// Deterministic fixed-order merge of the NBLK partials.
__global__ __launch_bounds__(256)
void magnn_reduce(const float* __restrict__ pm,
                  const float* __restrict__ ps,
                  const float* __restrict__ pacc,
                  float* __restrict__ out)              // (8*128)
{
    __shared__ float lds_M[HEADS];
    __shared__ float lds_S[HEADS];
    const int t = threadIdx.x;

    if (t < HEADS) {
        float M = NEG_BIG;
        for (int b = 0; b < NBLK; ++b) M = fmaxf(M, pm[b * HEADS + t]);
        lds_M[t] = M;
    }
    __syncthreads();
    if (t < HEADS) {
        const float M = lds_M[t];
        float S = 0.f;
        for (int b = 0; b < NBLK; ++b)
            S += __expf(pm[b * HEADS + t] - M) * ps[b * HEADS + t];
        lds_S[t] = S;
    }
    __syncthreads();

    const int h    = t >> 5;
    const int lane = t & 31;
    const float M  = lds_M[h];
    v4f acc = (v4f)(0.f);
    for (int b = 0; b < NBLK; ++b) {
        const float f = __expf(pm[b * HEADS + h] - M);
        const v4f   a = *(const v4f*)&pacc[(size_t)b * (HEADS * DIMD) + h * DIMD + lane * 4];
        acc += f * a;
    }
    const float invS = 1.0f / lds_S[h];
    acc *= invS;
    *(v4f*)&out[h * DIMD + lane * 4] = acc;
}

extern "C" void kernel_launch(void* const* d_in, const int* in_sizes, int n_in,
                              void* d_out, int out_size, void* d_ws, size_t ws_size,
                              hipStream_t stream) {
    const float* tgt   = (const float*)d_in[0];   // (128)
    const float* paths = (const float*)d_in[1];   // (100000,16,128)
    const float* afc   = (const float*)d_in[2];   // (8,256)
    float* out = (float*)d_out;                   // (1024)

    float* w    = (float*)d_ws;
    float* pm   = w;                              // NBLK*8
    float* ps   = w + NBLK * HEADS;               // NBLK*8
    float* pacc = w + 2 * NBLK * HEADS;           // NBLK*8*128

    magnn_pass1<<<NBLK, 256, 0, stream>>>(tgt, paths, afc, pm, ps, pacc);
    magnn_reduce<<<1, 256, 0, stream>>>(pm, ps, pacc, out);
}